// Attention_4183298146960
// MI455X (gfx1250) — compile-verified
//
#include <hip/hip_runtime.h>
#include <math.h>

typedef float v2f __attribute__((ext_vector_type(2)));
typedef float v8f __attribute__((ext_vector_type(8)));
typedef int v4i_vs __attribute__((vector_size(16)));  // matches builtin param

#define HEADS    16
#define KV_HEADS 4
#define HIDDEN   2048
#define HEAD_DIM 128
#define SEQ      2048
#define BATCH    2
#define KV_DIM   (KV_HEADS * HEAD_DIM)    // 512
#define NQKV     (HIDDEN + 2 * KV_DIM)    // 3072
#define MROWS    (BATCH * SEQ)            // 4096

// workspace layout (float elements)
#define WS_Q 0
#define WS_K (WS_Q + (size_t)BATCH * HEADS * SEQ * HEAD_DIM)
#define WS_V (WS_K + (size_t)BATCH * KV_HEADS * SEQ * HEAD_DIM)
#define WS_O (WS_V + (size_t)BATCH * KV_HEADS * SEQ * HEAD_DIM)

#if __has_builtin(__builtin_amdgcn_global_load_async_to_lds_b128)
#define HAVE_ASYNC 1
#else
#define HAVE_ASYNC 0
#endif

// 16-byte global -> LDS copy. Async (CDNA5 GLOBAL_LOAD_ASYNC_TO_LDS_B128,
// ASYNCcnt-tracked, no VGPR staging) when the builtin exists, else sync.
static __device__ __forceinline__ void cp16_to_lds(void* lds, const void* g) {
#if HAVE_ASYNC
  __builtin_amdgcn_global_load_async_to_lds_b128(
      (__attribute__((address_space(1))) v4i_vs*)(g),
      (__attribute__((address_space(3))) v4i_vs*)(lds), 0, 0);
#else
  *(float4*)lds = *(const float4*)g;
#endif
}

static __device__ __forceinline__ void wait_async_cp() {
#if HAVE_ASYNC
#if __has_builtin(__builtin_amdgcn_s_wait_asynccnt)
  __builtin_amdgcn_s_wait_asynccnt(0);
#else
  asm volatile("s_wait_asynccnt 0x0" ::: "memory");
#endif
#endif
}

static __device__ __forceinline__ v8f wmma_f32(v2f a, v2f b, v8f c) {
  return __builtin_amdgcn_wmma_f32_16x16x4_f32(false, a, false, b, (short)0, c,
                                               false, false);
}

static __device__ __forceinline__ float rope_rotate(float v, float partner,
                                                    int d, int pos) {
  // freq_i = 10000^(-2i/128) = exp(-i * ln(10000)/64)
  const float KLN = 0.14391156831f;  // ln(10000)/64
  float freq = expf(-KLN * (float)(d >> 1));
  float ang = (float)pos * freq;
  float sn, cs;
  sincosf(ang, &sn, &cs);
  // even: e*c - o*s ; odd: e*s + o*c
  return (d & 1) ? fmaf(partner, sn, v * cs) : fmaf(v, cs, -partner * sn);
}

// ---------------------------------------------------------------------------
// GEMM: C[M x N] = A[M x K] * Bw^T, A row-major, Bw is [N x K] row-major.
// Block tile 64(M) x 256(N), BK=16, double-buffered LDS with async copies.
// Tiles stored row-major with stride 20 floats: a fragment's (k,k+1) pair is
// one aligned ds_load_b64, and 20*j mod 64 makes 32-lane b64 reads hit all
// 64 banks (lane16-31 k+2 shift fills the odd pair slots) -> conflict-free.
// mode 0: plain store. mode 1: QKV epilogue (RoPE + scatter to [B,H,S,hd]).
// ---------------------------------------------------------------------------
#define BM 64
#define BN 256
#define BK 16
#define TSTR 20  // 16 + 4 pad

__global__ __launch_bounds__(256) void gemm_wmma_f32(
    const float* __restrict__ A, const float* __restrict__ Bw,
    float* __restrict__ outPlain, float* __restrict__ Qb,
    float* __restrict__ Kb, float* __restrict__ Vb, int K, int N, int mode) {
  __shared__ float As[2][BM * TSTR];
  __shared__ float Bs[2][BN * TSTR];

  const int t = threadIdx.x;
  const int lane = t & 31, w = t >> 5;
  const int hi = lane >> 4, lo = lane & 15;
  const int wm = w & 1, wn = w >> 1;
  const int rowBase = blockIdx.y * BM;
  const int colBase = blockIdx.x * BN;

  const v8f vzero = {0.f, 0.f, 0.f, 0.f, 0.f, 0.f, 0.f, 0.f};
  v8f acc[2][4];
#pragma unroll
  for (int i = 0; i < 2; ++i)
#pragma unroll
    for (int j = 0; j < 4; ++j) acc[i][j] = vzero;

  const int ar = t >> 2;        // 0..63 : A tile row
  const int ak = (t & 3) << 2;  // 0,4,8,12 : A tile k base

  auto load_tiles = [&](int buf, int kB) {
    cp16_to_lds(&As[buf][ar * TSTR + ak],
                A + (size_t)(rowBase + ar) * K + kB + ak);
    const float* bp = Bw + (size_t)(colBase + t) * K + kB;
#pragma unroll
    for (int i = 0; i < 4; ++i)
      cp16_to_lds(&Bs[buf][t * TSTR + i * 4], bp + i * 4);
  };

  load_tiles(0, 0);
  wait_async_cp();
  __syncthreads();

  int cur = 0;
  for (int kB = 0; kB < K; kB += BK) {
    if (kB + BK < K) load_tiles(cur ^ 1, kB + BK);  // prefetch next chunk

#pragma unroll
    for (int ks = 0; ks < BK; ks += 4) {
      const int kk = ks + 2 * hi;  // lanes 0-15: k,k+1 ; lanes 16-31: k+2,k+3
      v2f a[2], b[4];
#pragma unroll
      for (int mt = 0; mt < 2; ++mt) {
        const int m = wm * 32 + mt * 16 + lo;
        a[mt] = *(const v2f*)&As[cur][m * TSTR + kk];
      }
#pragma unroll
      for (int nt = 0; nt < 4; ++nt) {
        const int n = wn * 64 + nt * 16 + lo;
        b[nt] = *(const v2f*)&Bs[cur][n * TSTR + kk];
      }
#pragma unroll
      for (int mt = 0; mt < 2; ++mt)
#pragma unroll
        for (int nt = 0; nt < 4; ++nt)
          acc[mt][nt] = wmma_f32(a[mt], b[nt], acc[mt][nt]);
    }
    wait_async_cp();
    __syncthreads();
    cur ^= 1;
  }

  // epilogue
#pragma unroll
  for (int mt = 0; mt < 2; ++mt) {
#pragma unroll
    for (int nt = 0; nt < 4; ++nt) {
#pragma unroll
      for (int r = 0; r < 8; ++r) {
        const int row = rowBase + wm * 32 + mt * 16 + r + 8 * hi;
        const int col = colBase + wn * 64 + nt * 16 + lo;
        float v = acc[mt][nt][r];
        if (mode == 0) {
          outPlain[(size_t)row * N + col] = v;
        } else {
          const float partner = __shfl_xor(v, 1);
          const int bb = row >> 11, s = row & 2047;
          if (col < HIDDEN) {  // Q (rope)
            const int h = col >> 7, d = col & 127;
            Qb[(((size_t)bb * HEADS + h) * SEQ + s) * HEAD_DIM + d] =
                rope_rotate(v, partner, d, s);
          } else if (col < HIDDEN + KV_DIM) {  // K (rope)
            const int c = col - HIDDEN;
            const int h = c >> 7, d = c & 127;
            Kb[(((size_t)bb * KV_HEADS + h) * SEQ + s) * HEAD_DIM + d] =
                rope_rotate(v, partner, d, s);
          } else {  // V (raw)
            const int c = col - HIDDEN - KV_DIM;
            const int h = c >> 7, d = c & 127;
            Vb[(((size_t)bb * KV_HEADS + h) * SEQ + s) * HEAD_DIM + d] = v;
          }
        }
      }
    }
  }
}

// ---------------------------------------------------------------------------
// Flash attention: one workgroup per (qblock of 64 rows, head, batch).
// Online softmax; S and PV matmuls via v_wmma_f32_16x16x4_f32.
// K and V share one LDS buffer; V's (async) copy is issued before the softmax
// pass so the global->LDS transfer overlaps the exp/reduction work.
// ---------------------------------------------------------------------------
#define QSTR 132  // 128 + 4 pad; row stride 528B (16B aligned)
#define SSTR 66

__global__ __launch_bounds__(256) void attn_wmma_f32(
    const float* __restrict__ Qb, const float* __restrict__ Kb,
    const float* __restrict__ Vb, float* __restrict__ Ob) {
  __shared__ float Qs[64 * QSTR];
  __shared__ float KVs[64 * QSTR];
  __shared__ float Ss[64 * SSTR];
  __shared__ float AlphaS[64];
  __shared__ float Lrow[64];

  const int t = threadIdx.x;
  const int lane = t & 31, w = t >> 5;
  const int hi = lane >> 4, lo = lane & 15;
  const int qb = blockIdx.x, h = blockIdx.y, b = blockIdx.z;
  const int kvh = h >> 2;                    // GQA: 4 query heads per kv head
  const float scale = 0.08838834764831845f;  // 1/sqrt(128)

  const int smi = w >> 1, snb = (w & 1) * 2;  // S-phase wave mapping
  const int pmi = w & 3, pnq = w >> 2;        // PV-phase wave mapping

  // load Q tile [64 x 128] once (completion covered by first K-tile wait)
  {
    const float* qbase =
        Qb + (((size_t)b * HEADS + h) * SEQ + (size_t)qb * 64) * HEAD_DIM;
    for (int i = t; i < 512; i += 256) {
      const int r = i >> 3, c4 = i & 7;
      cp16_to_lds(&Qs[r * QSTR + c4 * 16], qbase + r * HEAD_DIM + c4 * 16);
    }
  }

  const v8f vzero = {0.f, 0.f, 0.f, 0.f, 0.f, 0.f, 0.f, 0.f};
  v8f oacc[4];
#pragma unroll
  for (int i = 0; i < 4; ++i) oacc[i] = vzero;
  float m_reg = -INFINITY, l_reg = 0.f;

  const int srow = t >> 2, sq4 = t & 3;  // softmax: 4 threads per row
  const int gq = qb * 64 + srow;         // global query row for this thread

  const float* kbase0 = Kb + ((size_t)b * KV_HEADS + kvh) * SEQ * HEAD_DIM;
  const float* vbase0 = Vb + ((size_t)b * KV_HEADS + kvh) * SEQ * HEAD_DIM;

  for (int kb = 0; kb <= qb; ++kb) {
    // ---- load K tile ----
    {
      const float* kbase = kbase0 + (size_t)kb * 64 * HEAD_DIM;
      for (int i = t; i < 512; i += 256) {
        const int r = i >> 3, c4 = i & 7;
        cp16_to_lds(&KVs[r * QSTR + c4 * 16], kbase + r * HEAD_DIM + c4 * 16);
      }
    }
    wait_async_cp();
    __syncthreads();

    // ---- S = scale * Q K^T  (64x64, K=128) ----
    {
      v8f sacc[2] = {vzero, vzero};
      const int m = smi * 16 + lo;
#pragma unroll
      for (int k = 0; k < 128; k += 4) {
        const int kk = k + 2 * hi;
        const v2f a = *(const v2f*)&Qs[m * QSTR + kk];
#pragma unroll
        for (int j = 0; j < 2; ++j) {
          const int n = (snb + j) * 16 + lo;
          const v2f bf = *(const v2f*)&KVs[n * QSTR + kk];
          sacc[j] = wmma_f32(a, bf, sacc[j]);
        }
      }
#pragma unroll
      for (int j = 0; j < 2; ++j)
#pragma unroll
        for (int r = 0; r < 8; ++r)
          Ss[(smi * 16 + r + 8 * hi) * SSTR + (snb + j) * 16 + lo] =
              sacc[j][r] * scale;
    }
    __syncthreads();

    // ---- issue V tile copy (reuses KVs), then online softmax on Ss ----
    {
      const float* vbase = vbase0 + (size_t)kb * 64 * HEAD_DIM;
      for (int i = t; i < 512; i += 256) {
        const int r = i >> 3, c4 = i & 7;
        cp16_to_lds(&KVs[r * QSTR + c4 * 16], vbase + r * HEAD_DIM + c4 * 16);
      }
    }
    {
      float sv[16];
      const int kc0 = kb * 64 + sq4 * 16;
      const bool diag = (kb == qb);
      float lmax = -INFINITY;
#pragma unroll
      for (int j = 0; j < 16; ++j) {
        float x = Ss[srow * SSTR + sq4 * 16 + j];
        if (diag && (kc0 + j) > gq) x = -INFINITY;  // causal mask
        sv[j] = x;
        lmax = fmaxf(lmax, x);
      }
      lmax = fmaxf(lmax, __shfl_xor(lmax, 1));
      lmax = fmaxf(lmax, __shfl_xor(lmax, 2));
      const float mnew = fmaxf(m_reg, lmax);
      const float alpha = expf(m_reg - mnew);
      float lsum = 0.f;
#pragma unroll
      for (int j = 0; j < 16; ++j) {
        const float p = expf(sv[j] - mnew);
        Ss[srow * SSTR + sq4 * 16 + j] = p;
        lsum += p;
      }
      lsum += __shfl_xor(lsum, 1);
      lsum += __shfl_xor(lsum, 2);
      l_reg = l_reg * alpha + lsum;
      m_reg = mnew;
      if (sq4 == 0) AlphaS[srow] = alpha;
    }
    wait_async_cp();
    __syncthreads();

    // ---- O = alpha*O + P V  (64x128, K=64) ----
    {
#pragma unroll
      for (int r = 0; r < 8; ++r) {
        const float al = AlphaS[pmi * 16 + r + 8 * hi];
#pragma unroll
        for (int nt = 0; nt < 4; ++nt) oacc[nt][r] *= al;
      }
      const int m = pmi * 16 + lo;
#pragma unroll
      for (int k = 0; k < 64; k += 4) {
        const int kk = k + 2 * hi;
        const v2f a = *(const v2f*)&Ss[m * SSTR + kk];
#pragma unroll
        for (int nt = 0; nt < 4; ++nt) {
          const int n = pnq * 64 + nt * 16 + lo;
          const v2f bf = v2f{KVs[kk * QSTR + n], KVs[(kk + 1) * QSTR + n]};
          oacc[nt] = wmma_f32(a, bf, oacc[nt]);
        }
      }
    }
    __syncthreads();
  }

  // ---- finalize: O /= l, write [B,S,HIDDEN] with col = h*128 + d ----
  if (sq4 == 0) Lrow[srow] = l_reg;
  __syncthreads();

  float* obase =
      Ob + ((size_t)b * SEQ + (size_t)qb * 64) * HIDDEN + h * HEAD_DIM;
#pragma unroll
  for (int r = 0; r < 8; ++r) {
    const int row = pmi * 16 + r + 8 * hi;
    const float linv = 1.0f / Lrow[row];
#pragma unroll
    for (int nt = 0; nt < 4; ++nt) {
      const int col = pnq * 64 + nt * 16 + lo;
      obase[(size_t)row * HIDDEN + col] = oacc[nt][r] * linv;
    }
  }
}

// ---------------------------------------------------------------------------
extern "C" void kernel_launch(void* const* d_in, const int* in_sizes, int n_in,
                              void* d_out, int out_size, void* d_ws,
                              size_t ws_size, hipStream_t stream) {
  const float* x = (const float*)d_in[0];     // [B,S,HIDDEN]
  const float* Wqkv = (const float*)d_in[1];  // [NQKV, HIDDEN]
  const float* Wo = (const float*)d_in[2];    // [HIDDEN, HIDDEN]
  float* out = (float*)d_out;                 // [B,S,HIDDEN]
  float* ws = (float*)d_ws;

  float* Qb = ws + WS_Q;
  float* Kb = ws + WS_K;
  float* Vb = ws + WS_V;
  float* Ob = ws + WS_O;

  // 1) QKV projection + RoPE + scatter to [B,H,S,hd]
  gemm_wmma_f32<<<dim3(NQKV / BN, MROWS / BM), 256, 0, stream>>>(
      x, Wqkv, nullptr, Qb, Kb, Vb, HIDDEN, NQKV, 1);

  // 2) causal flash attention -> Ob [B,S,HIDDEN]
  attn_wmma_f32<<<dim3(SEQ / 64, HEADS, BATCH), 256, 0, stream>>>(Qb, Kb, Vb,
                                                                  Ob);

  // 3) output projection
  gemm_wmma_f32<<<dim3(HIDDEN / BN, MROWS / BM), 256, 0, stream>>>(
      Ob, Wo, out, nullptr, nullptr, nullptr, HIDDEN, HIDDEN, 0);
}